// VocabParallelEmbeddingWithDelta_28973849379098
// MI455X (gfx1250) — compile-verified
//
#include <hip/hip_runtime.h>
#include <hip/hip_bf16.h>

// VocabParallelEmbeddingWithDelta: out[t,:] = weight[x[t],:] +
//   (indices[t] >= 0 ? delta_weights[indices[t], x[t], :] : 0)
//
// Pure HBM-streaming double gather (~320 MB traffic -> ~13.7 us floor at
// 23.3 TB/s; 1 add/element => no matrix-unit role). One 256-thread block
// (8 wave32) per token row:
//  - x[t], indices[t] -> uniform s_load; row bases in SGPRs
//  - straight-line body: each thread owns elements tid and tid+256 (no loop,
//    no per-lane exec masking)
//  - 128-bit coalesced loads/stores; non-temporal stores keep the 192MB L2
//    free for the gather tables (repeated tokens can hit in L2)

#define VOCAB 32000
#define EMBED 2048

typedef float v4f __attribute__((ext_vector_type(4)));

__global__ __launch_bounds__(256) void VocabParallelEmbeddingWithDelta_28973849379098_kernel(
    const int* __restrict__ x,
    const int* __restrict__ indices,
    const float* __restrict__ weight,
    const float* __restrict__ delta_weights,
    float* __restrict__ out,
    int num_tokens)
{
    const int t = blockIdx.x;
    if (t >= num_tokens) return;

    // Block-uniform scalar loads.
    const int tok = x[t];
    const int idx = indices[t];

    const v4f* __restrict__ wrow =
        (const v4f*)(weight + (size_t)tok * (size_t)EMBED);
    v4f* __restrict__ orow =
        (v4f*)(out + (size_t)t * (size_t)EMBED);

    const int i0 = threadIdx.x;        // 512 v4f per row; thread owns 2 of them
    const int i1 = threadIdx.x + 256;

    if (idx >= 0) {
        const v4f* __restrict__ drow =
            (const v4f*)(delta_weights +
                         ((size_t)idx * (size_t)VOCAB + (size_t)tok) * (size_t)EMBED);
        v4f a0 = wrow[i0];   // global_load_b128
        v4f a1 = wrow[i1];
        v4f b0 = drow[i0];
        v4f b1 = drow[i1];
        v4f r0 = a0 + b0;
        v4f r1 = a1 + b1;
        __builtin_nontemporal_store(r0, &orow[i0]);  // b128 NT store
        __builtin_nontemporal_store(r1, &orow[i1]);
    } else {
        // No adapter: copy base row.
        v4f a0 = wrow[i0];
        v4f a1 = wrow[i1];
        __builtin_nontemporal_store(a0, &orow[i0]);
        __builtin_nontemporal_store(a1, &orow[i1]);
    }
}

extern "C" void kernel_launch(void* const* d_in, const int* in_sizes, int n_in,
                              void* d_out, int out_size, void* d_ws, size_t ws_size,
                              hipStream_t stream) {
    const int*   xs      = (const int*)d_in[0];     // [T] token ids
    const int*   idxs    = (const int*)d_in[1];     // [T] adapter indices
    const float* weight  = (const float*)d_in[2];   // [VOCAB, EMBED]
    const float* deltas  = (const float*)d_in[3];   // [MAX_DELTAS, VOCAB, EMBED]
    float*       outp    = (float*)d_out;           // [T, EMBED]

    const int num_tokens = in_sizes[0];

    dim3 grid(num_tokens);
    dim3 block(256);  // 8 wave32 per block; 32B per thread per table
    VocabParallelEmbeddingWithDelta_28973849379098_kernel<<<grid, block, 0, stream>>>(
        xs, idxs, weight, deltas, outp, num_tokens);
}